// StochasticKeyNet_57509612094048
// MI455X (gfx1250) — compile-verified
//
#include <hip/hip_runtime.h>

// ---------------------------------------------------------------------------
// StochasticKeyNet forward for MI455X (gfx1250, wave32).
// All matmuls (9 convs as implicit GEMM + 2 FCs as 1x1 convs) run on the
// fp32 matrix pipe via V_WMMA_F32_16X16X4_F32 (exact fp32 semantics).
//
// GEMM mapping per layer:  D[M,N], M = Co, N = sp*128 + batch, K = taps x Cip.
//   - one block (1 wave) = 16(M) x 128(N) tile = one spatial position
//     -> padding checks are scalar-uniform; OOB taps skipped wholesale
//   - weights pre-packed to [Mpad][KH*KW][Cip] (zero pad) -> b64 A loads
//   - activations stored pair-interleaved: V'[(c/2)*HW+pix][b][c&1]
//     -> each B fragment is ONE coalesced b64 load; 8 staged fragments per
//        c-step give the scheduler MLP instead of load->wait0->wmma chains
// ---------------------------------------------------------------------------

typedef __attribute__((ext_vector_type(2))) float v2f;
typedef __attribute__((ext_vector_type(8))) float v8f;

#define BATCH 128
#define NSUB  8     // 8 x (16-wide N sub-tiles) = 128 columns per wave

// inv[perm[t]] = t  (argsort of a permutation == its inverse)
__global__ void invperm_kernel(const int* __restrict__ perm, int* __restrict__ inv, int n) {
    int t = blockIdx.x * blockDim.x + threadIdx.x;
    if (t < n) inv[perm[t]] = t;
}

// Decode x -> pair-interleaved V' and t-row buffer.
//   row < Ci*HW : Vp[((c>>1)*HW + pix)*256 + b*2 + (c&1)] = x[inv[row],b]/diag[inv[row]]
//   row == Ci*HW: T[b] = ...
__global__ void decode_kernel(const float* __restrict__ x, const int* __restrict__ inv,
                              const float* __restrict__ diag, float* __restrict__ vp,
                              float* __restrict__ T, int nrows, int logHW) {
    int tid = blockIdx.x * blockDim.x + threadIdx.x;
    if (tid >= nrows * BATCH) return;
    int row = tid >> 7;
    int b   = tid & 127;
    int j = inv[row];
    float val = x[j * BATCH + b] / diag[j];
    if (row == nrows - 1) {
        T[b] = val;
    } else {
        int c   = row >> logHW;
        int pix = row & ((1 << logHW) - 1);
        size_t idx = (((size_t)(c >> 1) << logHW) + pix) * 256 + (b << 1) + (c & 1);
        vp[idx] = val;
    }
}

// xn[j, b] = maybe_relu(diag[j] * yaug[perm[j], b]);  yaug row Mout == t row
__global__ void encode_kernel(const int* __restrict__ perm, const float* __restrict__ diag,
                              const float* __restrict__ y, const float* __restrict__ T,
                              int Mout, float* __restrict__ xn, int nrows, int do_relu) {
    int tid = blockIdx.x * blockDim.x + threadIdx.x;
    if (tid >= nrows * BATCH) return;
    int j = tid >> 7;
    int b = tid & 127;
    int p = perm[j];
    float val = (p < Mout) ? y[(size_t)p * BATCH + b] : T[b];
    val *= diag[j];
    if (do_relu) val = fmaxf(val, 0.0f);
    xn[tid] = val;
}

// wpack[m][tap][c] = (m<Co && c<Ci) ? w[m][c][tap] : 0   (w is (Co,Ci,KH,KW) row-major)
__global__ void pack_weights_kernel(const float* __restrict__ w, float* __restrict__ wpack,
                                    int Co, int Ci, int KK, int Cip, int total) {
    int idx = blockIdx.x * blockDim.x + threadIdx.x;
    if (idx >= total) return;
    int c    = idx % Cip;
    int rest = idx / Cip;
    int tap  = rest % KK;
    int m    = rest / KK;
    wpack[idx] = (m < Co && c < Ci) ? w[((size_t)m * Ci + c) * KK + tap] : 0.0f;
}

// Implicit-GEMM conv (FC == 1x1 conv).  One wave -> 16(M) x 128(N) D tile.
// ISA VGPR layouts (CDNA5 ISA §7.12.2):
//   A 16x4 f32 : lanes 0-15 K={0,1} in v[0],v[1]; lanes 16-31 K={2,3}
//   B 4x16 f32 : lanes 0-15 K rows {0,1};         lanes 16-31 K rows {2,3}
//   C/D 16x16  : VGPR r -> M=r (lanes 0-15) / M=r+8 (lanes 16-31), N=lane%16
__global__ __launch_bounds__(32) void conv_gemm_wmma(
    const float* __restrict__ vp,    // pair-interleaved activations (Cip/2, HW, 128, 2)
    const float* __restrict__ T,     // t row (128)
    const float* __restrict__ wp,    // Mpad x KK x Cip packed weights (zero padded)
    const float* __restrict__ bias,  // Co
    float* __restrict__ y,           // (Co*Ho*Wo) x 128
    int H, int W, int Co, int KH, int KW, int stride, int pad,
    int Wo, int HoWo, int Cip)
{
    const int lane = threadIdx.x;
    const int half = lane >> 4;    // 0: lanes 0-15, 1: lanes 16-31
    const int l16  = lane & 15;
    const int HW   = H * W;
    const int KK   = KH * KW;

    const int sp    = blockIdx.x;          // output spatial position (uniform!)
    const int mtile = blockIdx.y * 16;
    const int oh = sp / Wo, ow = sp - oh * Wo;
    const int ih0 = oh * stride - pad;
    const int iw0 = ow * stride - pad;

    v8f acc[NSUB];
    const v8f vzero = {0.f, 0.f, 0.f, 0.f, 0.f, 0.f, 0.f, 0.f};
#pragma unroll
    for (int s = 0; s < NSUB; ++s) acc[s] = vzero;

    // A row pointer for this lane: wp[(mtile+l16)][tap=0][half*2]
    const float* arow = wp + (size_t)(mtile + l16) * ((size_t)KK * Cip) + (half << 1);
    const size_t pairStride = (size_t)HW << 8;   // floats per channel-pair plane

    for (int kh = 0; kh < KH; ++kh) {
        const int ih = ih0 + kh;
        for (int kw = 0; kw < KW; ++kw, arow += Cip) {
            const int iw = iw0 + kw;
            // Uniform (scalar) padding test: OOB tap contributes exactly zero.
            if ((unsigned)ih >= (unsigned)H || (unsigned)iw >= (unsigned)W) continue;

            // per-lane base: pair plane 'half', pixel (ih,iw), column l16
            const float* bbase = vp + ((size_t)(ih * W + iw) << 8) + (l16 << 1)
                                    + (size_t)half * pairStride;
            for (int c = 0; c < Cip; c += 4) {
                const v2f a = *(const v2f*)(arow + c);                 // b64 A load
                const float* bp = bbase + (size_t)(c >> 1) * pairStride;
                v2f bf[NSUB];
#pragma unroll
                for (int s = 0; s < NSUB; ++s) {                       // 8 staged b64 loads
                    bf[s] = *(const v2f*)(bp + s * 32);
                }
#pragma unroll
                for (int s = 0; s < NSUB; ++s) {
                    acc[s] = __builtin_amdgcn_wmma_f32_16x16x4_f32(
                        false, a, false, bf[s], (short)0, acc[s], false, false);
                }
            }
        }
    }

    // Epilogue: y[mr, sp, b] = acc + bias[mr] * t[b]
#pragma unroll
    for (int s = 0; s < NSUB; ++s) {
        const int bb = s * 16 + l16;
        const float tb = T[bb];
#pragma unroll
        for (int r = 0; r < 8; ++r) {
            const int mr = mtile + r + (half << 3);
            if (mr < Co) {
                y[((size_t)mr * HoWo + sp) * BATCH + bb] = acc[s][r] + bias[mr] * tb;
            }
        }
    }
}

extern "C" void kernel_launch(void* const* d_in, const int* in_sizes, int n_in,
                              void* d_out, int out_size, void* d_ws, size_t ws_size,
                              hipStream_t stream)
{
    (void)in_sizes; (void)n_in; (void)out_size; (void)ws_size;

    const float* x0 = (const float*)d_in[0];
    const int*   perm[12];
    const float* diag[12];
    for (int i = 0; i < 12; ++i) {
        perm[i] = (const int*)  d_in[1 + 2 * i];
        diag[i] = (const float*)d_in[2 + 2 * i];
    }
    const float *cw[9], *cb[9];
    for (int j = 0; j < 9; ++j) {
        cw[j] = (const float*)d_in[25 + 2 * j];
        cb[j] = (const float*)d_in[26 + 2 * j];
    }
    const float* fw1 = (const float*)d_in[43];
    const float* fb1 = (const float*)d_in[44];
    const float* fw2 = (const float*)d_in[45];
    const float* fb2 = (const float*)d_in[46];

    static const int nrows[12] = {3073, 98305, 98305, 24577, 49153, 49153,
                                  12289, 12289, 12289, 641, 101, 11};

    // Workspace carve-out (~154 MB): V' (pair-interleaved), Y, X ping, weights, T, inverses.
    char* ws = (char*)d_ws;
    size_t off = 0;
    auto take = [&](size_t bytes) -> void* {
        void* p = ws + off;
        off = (off + bytes + 255) & ~(size_t)255;
        return p;
    };
    float* V  = (float*)take((size_t)48 * 1024 * 256 * sizeof(float));  // max (Cip/2)*HW*256
    float* Y  = (float*)take((size_t)96 * 1024 * BATCH * sizeof(float));
    float* X  = (float*)take((size_t)98305 * BATCH * sizeof(float));
    float* WPACK = (float*)take((size_t)192 * 9 * 192 * sizeof(float));
    float* T  = (float*)take((size_t)BATCH * sizeof(float));
    int* inv[12];
    for (int i = 0; i < 12; ++i) inv[i] = (int*)take((size_t)nrows[i] * sizeof(int));

    // Build inverse permutations (argsort of a permutation).
    for (int i = 0; i < 12; ++i) {
        invperm_kernel<<<(nrows[i] + 255) / 256, 256, 0, stream>>>(perm[i], inv[i], nrows[i]);
    }

    struct LP { int Ci, H, W, Co, KH, KW, st, pad, Ho, Wo, logHW; const float *w, *b; };
    const LP L[11] = {
        {  3, 32, 32,  96, 3, 3, 1, 1, 32, 32, 10, cw[0], cb[0]},
        { 96, 32, 32,  96, 3, 3, 1, 1, 32, 32, 10, cw[1], cb[1]},
        { 96, 32, 32,  96, 3, 3, 2, 1, 16, 16, 10, cw[2], cb[2]},
        { 96, 16, 16, 192, 3, 3, 1, 1, 16, 16,  8, cw[3], cb[3]},
        {192, 16, 16, 192, 3, 3, 1, 1, 16, 16,  8, cw[4], cb[4]},
        {192, 16, 16, 192, 3, 3, 2, 1,  8,  8,  8, cw[5], cb[5]},
        {192,  8,  8, 192, 3, 3, 1, 1,  8,  8,  6, cw[6], cb[6]},
        {192,  8,  8, 192, 1, 1, 1, 0,  8,  8,  6, cw[7], cb[7]},
        {192,  8,  8,  10, 1, 1, 1, 0,  8,  8,  6, cw[8], cb[8]},
        {640,  1,  1, 100, 1, 1, 1, 0,  1,  1,  0, fw1,  fb1 },   // fc1 as 1x1 conv
        {100,  1,  1,  10, 1, 1, 1, 0,  1,  1,  0, fw2,  fb2 },   // fc2 as 1x1 conv
    };

    const float* xcur = x0;
    for (int l = 0; l < 11; ++l) {
        const LP& p = L[l];
        const int n_in_rows  = p.Ci * p.H * p.W + 1;   // == nrows[l]
        const int Mout       = p.Co * p.Ho * p.Wo;
        const int n_out_rows = Mout + 1;               // == nrows[l+1]
        const int KK   = p.KH * p.KW;
        const int Cip  = (p.Ci + 3) & ~3;              // channel pad to K-step
        const int Mpad = (p.Co + 15) & ~15;            // M pad to WMMA tile
        const int HoWo = p.Ho * p.Wo;

        // decode: x -> pair-interleaved V' + T row
        {
            int tot = n_in_rows * BATCH;
            decode_kernel<<<(tot + 255) / 256, 256, 0, stream>>>(
                xcur, inv[l], diag[l], V, T, n_in_rows, p.logHW);
        }

        // pack weights: (Co,Ci,KH,KW) -> [Mpad][KK][Cip], zero padded
        {
            int tot = Mpad * KK * Cip;
            pack_weights_kernel<<<(tot + 255) / 256, 256, 0, stream>>>(
                p.w, WPACK, p.Co, p.Ci, KK, Cip, tot);
        }

        // conv/FC as fp32 WMMA implicit GEMM: V' -> Y
        dim3 grid(HoWo, Mpad / 16);
        conv_gemm_wmma<<<grid, 32, 0, stream>>>(V, T, WPACK, p.b, Y,
            p.H, p.W, p.Co, p.KH, p.KW, p.st, p.pad, p.Wo, HoWo, Cip);

        // encode(+ReLU except final): [Y ; t] -> next x (or d_out on last layer)
        float* xn = (l == 10) ? (float*)d_out : X;
        int tot = n_out_rows * BATCH;
        encode_kernel<<<(tot + 255) / 256, 256, 0, stream>>>(
            perm[l + 1], diag[l + 1], Y, T, Mout, xn, n_out_rows, (l < 10) ? 1 : 0);

        xcur = xn;
    }
}